// BrightnessSampler_62912680952008
// MI455X (gfx1250) — compile-verified
//
#include <hip/hip_runtime.h>

// ---------------------------------------------------------------------------
// BrightnessSampler for MI455X (gfx1250, wave32)
//   score GEMM  : v_wmma_f32_16x16x4_f32  (K=3 padded to 4)
//   ranking     : per-row counting sort (ballot/popcount wave primitives)
//   streaming   : float4 non-temporal copies for the samp passthrough
// ---------------------------------------------------------------------------

typedef float v2f __attribute__((ext_vector_type(2)));
typedef float v4f __attribute__((ext_vector_type(4)));
typedef float v8f __attribute__((ext_vector_type(8)));

#define N_SEL   128
#define M_CACHE 384
#define B_TOT   65536
#define ROWS    16          // b-rows per block (one WMMA tile of rows)
#define NBUCK   256

__device__ __forceinline__ unsigned ballot32(bool p) {
  return __builtin_amdgcn_ballot_w32(p);
}
// monotone float -> orderable uint (total order, -0 < +0)
__device__ __forceinline__ unsigned keyOf(float f) {
  unsigned u = __float_as_uint(f);
  unsigned m = (u & 0x80000000u) ? 0xFFFFFFFFu : 0x80000000u;
  return u ^ m;
}
__device__ __forceinline__ int bucketOf(float s) {
  int bk = (int)((s + 1.0f) * 128.0f);       // uniform: dot of random unit vecs
  return bk < 0 ? 0 : (bk > 255 ? 255 : bk);
}

__global__ __launch_bounds__(256) void brightness_sampler_kernel(
    const float* __restrict__ cache,    // (384,5)
    const float* __restrict__ refdirs,  // (B,3)
    const float* __restrict__ normal,   // (B,3)
    const float* __restrict__ sampRays, // (B,128,3)
    const float* __restrict__ sampMip,  // (B,128)
    float* __restrict__ out)            // rays (B,256,3) ++ mip (B,256)
{
  __shared__ v4f            sCache[M_CACHE];          // dir.xyz + mip
  __shared__ float          sScore[ROWS * M_CACHE];
  __shared__ unsigned       sHist[ROWS * NBUCK];
  __shared__ unsigned short sIdx[ROWS * M_CACHE];

  const int tid   = threadIdx.x;
  const int lane  = tid & 31;
  const int w     = tid >> 5;
  const int bbase = blockIdx.x * ROWS;

  float* raysOut = out;
  float* mipOut  = out + (size_t)B_TOT * 768;

  // ---- phase 0: streaming pass-through copy (non-temporal) ----------------
  {
    const v4f* srcR = (const v4f*)sampRays;
    v4f*       dstR = (v4f*)raysOut;
    for (int i = tid; i < ROWS * 96; i += 256) {       // 96 f4 per row
      int r = i / 96, q = i - r * 96;
      v4f v = __builtin_nontemporal_load(&srcR[(size_t)(bbase + r) * 96 + q]);
      __builtin_nontemporal_store(v, &dstR[(size_t)(bbase + r) * 192 + q]);
    }
    const v4f* srcM = (const v4f*)sampMip;
    v4f*       dstM = (v4f*)mipOut;
    for (int i = tid; i < ROWS * 32; i += 256) {       // 32 f4 per row
      int r = i >> 5, q = i & 31;
      v4f v = __builtin_nontemporal_load(&srcM[(size_t)(bbase + r) * 32 + q]);
      __builtin_nontemporal_store(v, &dstM[(size_t)(bbase + r) * 64 + q]);
    }
  }

  // ---- phase 1: cache -> LDS, zero histograms -----------------------------
  for (int m = tid; m < M_CACHE; m += 256) {
    const float* cp = cache + m * 5;
    v4f c; c.x = cp[0]; c.y = cp[1]; c.z = cp[2]; c.w = cp[3];
    sCache[m] = c;
  }
  for (int i = tid; i < ROWS * NBUCK; i += 256) sHist[i] = 0u;
  __syncthreads();

  // ---- phase 2: WMMA score GEMM (wave w owns column tiles 3w..3w+2) -------
  {
    const int li = lane & 15, hf = lane >> 4;
    const size_t brow = (size_t)(bbase + li);
    const float* np = normal  + brow * 3;
    const float* rp = refdirs + brow * 3;
    // A 16x4 f32 layout: lanes 0-15 hold K0/K1, lanes 16-31 hold K2/K3 (K3=0 pad)
    v2f aN, aR;
    aN.x = hf ? np[2] : np[0]; aN.y = hf ? 0.f : np[1];
    aR.x = hf ? rp[2] : rp[0]; aR.y = hf ? 0.f : rp[1];
    v8f z = {};
    for (int t = w * 3; t < w * 3 + 3; ++t) {
      int m = t * 16 + li;
      v4f c = sCache[m];
      v2f bT; bT.x = hf ? c.z : c.x; bT.y = hf ? 0.f : c.y;  // B 4x16, K3=0 pad
      v8f hemi = __builtin_amdgcn_wmma_f32_16x16x4_f32(
          false, aN, false, bT, (short)0, z, false, false);
      v8f sim  = __builtin_amdgcn_wmma_f32_16x16x4_f32(
          false, aR, false, bT, (short)0, z, false, false);
#pragma unroll
      for (int i = 0; i < 8; ++i) {                      // D: row = i + 8*hf
        float sc = (hemi[i] > 0.f) ? -sim[i] : 0.f;
        sScore[(i + hf * 8) * M_CACHE + m] = sc;
      }
    }
  }
  __syncthreads();

  // ---- phase 3: per-row counting-sort ranking (wave w -> rows 2w,2w+1) ----
  const unsigned laneLt = (1u << lane) - 1u;
  for (int rr = 0; rr < 2; ++rr) {
    const int row  = w * 2 + rr;
    const int brow = bbase + row;
    const float*    srow = &sScore[row * M_CACHE];
    unsigned*       hrow = &sHist[row * NBUCK];
    unsigned short* sidx = &sIdx[row * M_CACHE];

    // classify: neg (sorted first), zero (index order), pos (sorted last)
    float sv[12]; unsigned bz[12];
    unsigned nNeg = 0, nZero = 0;
#pragma unroll
    for (int k = 0; k < 12; ++k) {
      sv[k] = srow[k * 32 + lane];
      unsigned bn = ballot32(sv[k] < 0.f);
      bz[k] = ballot32(sv[k] == 0.f);
      nNeg += __popc(bn); nZero += __popc(bz[k]);
    }

    // histogram of nonzeros
#pragma unroll
    for (int k = 0; k < 12; ++k)
      if (sv[k] != 0.f) atomicAdd(&hrow[bucketOf(sv[k])], 1u);
    __threadfence_block();

    // exclusive scan over 256 buckets (8 per lane + wave scan)
    unsigned c[8], lt = 0;
#pragma unroll
    for (int q = 0; q < 8; ++q) { c[q] = hrow[lane * 8 + q]; lt += c[q]; }
    unsigned incl = lt;
    for (int d = 1; d < 32; d <<= 1) {
      unsigned n = (unsigned)__shfl_up((int)incl, d, 32);
      if (lane >= d) incl += n;
    }
    unsigned run = incl - lt;
#pragma unroll
    for (int q = 0; q < 8; ++q) { unsigned cc = c[q]; hrow[lane * 8 + q] = run; run += cc; }
    __threadfence_block();

    // counting-sort scatter of nonzero element indices
#pragma unroll
    for (int k = 0; k < 12; ++k)
      if (sv[k] != 0.f) {
        unsigned pos = atomicAdd(&hrow[bucketOf(sv[k])], 1u);
        sidx[pos] = (unsigned short)(k * 32 + lane);
      }
    __threadfence_block();

    // intra-bucket fix: insertion sort by (key, idx); segments are tiny
#pragma unroll 1
    for (int q = 0; q < 8; ++q) {
      int bk = lane * 8 + q;
      unsigned end   = hrow[bk];
      unsigned start = bk ? hrow[bk - 1] : 0u;
      for (unsigned i = start + 1; i < end; ++i) {
        unsigned short vi = sidx[i];
        unsigned kv = keyOf(srow[vi]);
        unsigned jj = i;
        while (jj > start) {
          unsigned short vp = sidx[jj - 1];
          unsigned kp = keyOf(srow[vp]);
          if (kp < kv || (kp == kv && vp < vi)) break;
          sidx[jj] = vp; --jj;
        }
        sidx[jj] = vi;
      }
    }
    __threadfence_block();

    // compaction: selected = original index < 128; slot = output position,
    // rank = position in the full stable order (= gather index per reference)
    unsigned nNonzero = M_CACHE - nZero;
    unsigned bs[12];
#pragma unroll
    for (int k = 0; k < 12; ++k) {
      unsigned p = k * 32 + lane;
      bool val = p < nNonzero;
      int fi = val ? (int)sidx[p] : 0x7FFF;
      bs[k] = ballot32(val && (fi < N_SEL));
    }
    unsigned kb = nNeg >> 5, rb = nNeg & 31;
    unsigned nNegSel = 0;
#pragma unroll
    for (int k = 0; k < 12; ++k) if ((unsigned)k < kb) nNegSel += __popc(bs[k]);
    if (kb < 12u && rb) nNegSel += __popc(bs[kb] & ((1u << rb) - 1u));
    unsigned nZeroSel = __popc(bz[0]) + __popc(bz[1]) + __popc(bz[2]) + __popc(bz[3]);

    const size_t rayBase = (size_t)brow * 768 + 384;
    const size_t mipBase = (size_t)brow * 256 + 128;

    unsigned acc = 0;
#pragma unroll
    for (int k = 0; k < 12; ++k) {                 // selected nonzeros
      unsigned ball = bs[k];
      unsigned pre  = acc + __popc(ball & laneLt);
      if ((ball >> lane) & 1u) {
        unsigned p    = k * 32 + lane;
        unsigned rank = (p < nNeg) ? p   : p + nZero;
        unsigned slot = (p < nNeg) ? pre : pre + nZeroSel;
        v4f cv = sCache[rank];
        raysOut[rayBase + slot * 3 + 0] = cv.x;
        raysOut[rayBase + slot * 3 + 1] = cv.y;
        raysOut[rayBase + slot * 3 + 2] = cv.z;
        mipOut[mipBase + slot] = cv.w;
      }
      acc += __popc(ball);
    }
    unsigned zacc = 0;
#pragma unroll
    for (int k = 0; k < 4; ++k) {                  // selected zeros (m < 128)
      unsigned ball = bz[k];
      unsigned zp   = zacc + __popc(ball & laneLt);
      if ((ball >> lane) & 1u) {
        unsigned rank = nNeg + zp;
        unsigned slot = nNegSel + zp;
        v4f cv = sCache[rank];
        raysOut[rayBase + slot * 3 + 0] = cv.x;
        raysOut[rayBase + slot * 3 + 1] = cv.y;
        raysOut[rayBase + slot * 3 + 2] = cv.z;
        mipOut[mipBase + slot] = cv.w;
      }
      zacc += __popc(ball);
    }
  }
}

extern "C" void kernel_launch(void* const* d_in, const int* in_sizes, int n_in,
                              void* d_out, int out_size, void* d_ws, size_t ws_size,
                              hipStream_t stream) {
  (void)in_sizes; (void)n_in; (void)out_size; (void)d_ws; (void)ws_size;
  const float* cache    = (const float*)d_in[0];
  const float* refdirs  = (const float*)d_in[1];
  const float* normal   = (const float*)d_in[2];
  const float* sampRays = (const float*)d_in[3];
  const float* sampMip  = (const float*)d_in[4];
  float* out = (float*)d_out;
  dim3 grid(B_TOT / ROWS), block(256);
  brightness_sampler_kernel<<<grid, block, 0, stream>>>(
      cache, refdirs, normal, sampRays, sampMip, out);
}